// GraphTransformerLayer_42030549958707
// MI455X (gfx1250) — compile-verified
//
#include <hip/hip_runtime.h>
#include <math.h>

#define DM   128
#define DFF  256
#define LDA  132   // 128 + 4 pad (stride mod 64 banks = 4 -> conflict-free b64 LDS reads)
#define LDY  260   // 256 + 4 pad

typedef __attribute__((ext_vector_type(2))) float v2f;
typedef __attribute__((ext_vector_type(8))) float v8f;

// ---- CDNA5 WMMA: D = A(16x4 f32) * B(4x16 f32) + C(16x16 f32) -------------
__device__ __forceinline__ v8f wmma4(v2f a, v2f b, v8f c) {
  return __builtin_amdgcn_wmma_f32_16x16x4_f32(false, a, false, b, (short)0, c,
                                               false, false);
}

// Accumulate a full 16x16 tile over K (K multiple of 4).
// arow = &A[lo*lda + 2*hi]  (A row for this lane, pre-offset by K sub-phase)
// brow = &B[(n0+lo)*ldb + 2*hi]  where GEMM is  X @ W^T  (B[k][n] = W[n][k])
template <int K>
__device__ __forceinline__ v8f gemm_k(const float* arow, const float* brow) {
  v8f acc = {};
#pragma unroll
  for (int k0 = 0; k0 < K; k0 += 4) {
    v2f a = *(const v2f*)(arow + k0);
    v2f b = *(const v2f*)(brow + k0);
    acc = wmma4(a, b, acc);
  }
  return acc;
}

__device__ __forceinline__ void atomicAddF(float* p, float v) {
  unsafeAtomicAdd(p, v);   // lowers to global_atomic_add_f32 on gfx1250
}

// ---------------------------------------------------------------------------
__global__ __launch_bounds__(256) void zero_kernel(float* p, int n) {
  int i = blockIdx.x * 256 + threadIdx.x;
  if (i < n) p[i] = 0.0f;
}

// ---- Kernel 1: Q/K/V projections (h @ W^T), WMMA f32 ----------------------
__global__ __launch_bounds__(256) void qkv_kernel(
    const float* __restrict__ h, const float* __restrict__ Wq,
    const float* __restrict__ Wk, const float* __restrict__ Wv,
    float* __restrict__ Q, float* __restrict__ K, float* __restrict__ V, int N) {
  __shared__ float hs[16 * LDA];
  const int t = threadIdx.x;
  const int node0 = blockIdx.x * 16;
  for (int i = t; i < 16 * DM; i += 256) {
    int r = i >> 7, c = i & 127;
    int node = node0 + r;
    hs[r * LDA + c] = (node < N) ? h[node * DM + c] : 0.0f;
  }
  __syncthreads();

  const int w = t >> 5, lane = t & 31, lo = lane & 15, hi = lane >> 4;
  const float* arow = &hs[lo * LDA + 2 * hi];
  const int n0 = w * 16;

  const float* mats[3] = {Wq, Wk, Wv};
  float* outs[3] = {Q, K, V};
#pragma unroll
  for (int m = 0; m < 3; ++m) {
    const float* brow = mats[m] + (n0 + lo) * DM + 2 * hi;
    v8f acc = gemm_k<DM>(arow, brow);
    float* o = outs[m];
#pragma unroll
    for (int i = 0; i < 8; ++i) {
      int node = node0 + i + 8 * hi;
      if (node < N) o[node * DM + n0 + lo] = acc[i];
    }
  }
}

// ---- Kernel 2: per-edge attention, scatter-add (bandwidth-bound) ----------
__global__ __launch_bounds__(256) void edge_attn_kernel(
    const float* __restrict__ Q, const float* __restrict__ K,
    const float* __restrict__ V, const int* __restrict__ src,
    const int* __restrict__ dst, float* __restrict__ wV, float* __restrict__ z,
    int E) {
  int idx = blockIdx.x * 256 + threadIdx.x;
  int e = idx >> 7;
  if (e >= E) return;
  int c = idx & 127;           // channel = head*16 + d
  int head = c >> 4;
  int s = src[e], d = dst[e];

  const float* vptr = &V[s * DM + c];
  __builtin_prefetch(vptr, 0, 0);          // global_prefetch_b8

  float p = K[s * DM + c] * Q[d * DM + c];
#pragma unroll
  for (int off = 8; off > 0; off >>= 1) p += __shfl_xor(p, off, 16);

  float score = fminf(5.0f, fmaxf(-5.0f, p * 0.25f));  // 1/sqrt(16)
  float sv = __expf(score);
  atomicAddF(&wV[d * DM + c], vptr[0] * sv);
  if ((c & 15) == 0) atomicAddF(&z[d * 8 + head], sv);
}

// ---- Kernel 3: x = (wV/z) @ Wo^T + bo + h ---------------------------------
__global__ __launch_bounds__(256) void attn_out_kernel(
    const float* __restrict__ wV, const float* __restrict__ z,
    const float* __restrict__ h, const float* __restrict__ Wo,
    const float* __restrict__ bo, float* __restrict__ x, int N) {
  __shared__ float as[16 * LDA];
  const int t = threadIdx.x;
  const int node0 = blockIdx.x * 16;
  for (int i = t; i < 16 * DM; i += 256) {
    int r = i >> 7, c = i & 127;
    int node = node0 + r;
    float v = 0.0f;
    if (node < N) v = wV[node * DM + c] / (z[node * 8 + (c >> 4)] + 1e-6f);
    as[r * LDA + c] = v;
  }
  __syncthreads();

  const int w = t >> 5, lane = t & 31, lo = lane & 15, hi = lane >> 4;
  const float* arow = &as[lo * LDA + 2 * hi];
  const int n0 = w * 16;
  const float* brow = Wo + (n0 + lo) * DM + 2 * hi;
  v8f acc = gemm_k<DM>(arow, brow);
#pragma unroll
  for (int i = 0; i < 8; ++i) {
    int node = node0 + i + 8 * hi;
    int col = n0 + lo;
    if (node < N) x[node * DM + col] = acc[i] + bo[col] + h[node * DM + col];
  }
}

// ---- Column sums / sums-of-squares for BatchNorm stats --------------------
__global__ __launch_bounds__(256) void col_stats_kernel(
    const float* __restrict__ X, float* __restrict__ stats, int N) {
  __shared__ float s1[256], s2[256];
  const int t = threadIdx.x;
  const int j = t & 127, r = t >> 7;
  const int base = blockIdx.x * 256;
  float a = 0.0f, b = 0.0f;
  for (int rr = r; rr < 256; rr += 2) {
    int n = base + rr;
    if (n < N) {
      float v = X[n * DM + j];
      a += v;
      b += v * v;
    }
  }
  s1[t] = a; s2[t] = b;
  __syncthreads();
  if (t < 128) {
    atomicAddF(&stats[j],       s1[t] + s1[t + 128]);
    atomicAddF(&stats[128 + j], s2[t] + s2[t + 128]);
  }
}

// ---- Kernel 4: BN1-normalize, FFN (128->256 relu ->128) + residual --------
__global__ __launch_bounds__(256) void ffn_kernel(
    const float* __restrict__ x, const float* __restrict__ stats1,
    const float* __restrict__ bn1_w, const float* __restrict__ bn1_b,
    const float* __restrict__ W1, const float* __restrict__ b1,
    const float* __restrict__ W2, const float* __restrict__ b2,
    float* __restrict__ ypre, float invN, int N) {
  __shared__ float xs[16 * LDA];
  __shared__ float ys[16 * LDY];
  const int t = threadIdx.x;
  const int node0 = blockIdx.x * 16;
  for (int i = t; i < 16 * DM; i += 256) {
    int r = i >> 7, c = i & 127;
    int node = node0 + r;
    float m = stats1[c] * invN;
    float var = stats1[128 + c] * invN - m * m;
    float xv = (node < N) ? x[node * DM + c] : 0.0f;
    xs[r * LDA + c] = (xv - m) * rsqrtf(var + 1e-5f) * bn1_w[c] + bn1_b[c];
  }
  __syncthreads();

  const int w = t >> 5, lane = t & 31, lo = lane & 15, hi = lane >> 4;
  const float* arow = &xs[lo * LDA + 2 * hi];

  // phase 1: y1 = relu(xn @ W1^T + b1), wave w covers cols [32w, 32w+32)
#pragma unroll
  for (int tt = 0; tt < 2; ++tt) {
    int n0 = w * 32 + tt * 16;
    const float* brow = W1 + (n0 + lo) * DM + 2 * hi;
    v8f acc = gemm_k<DM>(arow, brow);
#pragma unroll
    for (int i = 0; i < 8; ++i) {
      int rr = i + 8 * hi, col = n0 + lo;
      ys[rr * LDY + col] = fmaxf(acc[i] + b1[col], 0.0f);
    }
  }
  __syncthreads();

  // phase 2: y = y1 @ W2^T + b2 + xn, wave w covers cols [16w, 16w+16)
  {
    int n0 = w * 16;
    const float* arow2 = &ys[lo * LDY + 2 * hi];
    const float* brow = W2 + (n0 + lo) * DFF + 2 * hi;
    v8f acc = gemm_k<DFF>(arow2, brow);
#pragma unroll
    for (int i = 0; i < 8; ++i) {
      int rr = i + 8 * hi, col = n0 + lo;
      int node = node0 + rr;
      if (node < N)
        ypre[node * DM + col] = acc[i] + b2[col] + xs[rr * LDA + col];
    }
  }
}

// ---- Kernel 5: apply BN2 -> output ----------------------------------------
__global__ __launch_bounds__(256) void bn_apply_kernel(
    const float* __restrict__ ypre, const float* __restrict__ stats2,
    const float* __restrict__ w, const float* __restrict__ b,
    float* __restrict__ out, float invN, int total) {
  int i = blockIdx.x * 256 + threadIdx.x;
  if (i >= total) return;
  int j = i & 127;
  float m = stats2[j] * invN;
  float var = stats2[128 + j] * invN - m * m;
  out[i] = (ypre[i] - m) * rsqrtf(var + 1e-5f) * w[j] + b[j];
}

// ---------------------------------------------------------------------------
extern "C" void kernel_launch(void* const* d_in, const int* in_sizes, int n_in,
                              void* d_out, int out_size, void* d_ws,
                              size_t ws_size, hipStream_t stream) {
  const float* h    = (const float*)d_in[0];
  const int*   src  = (const int*)d_in[1];
  const int*   dst  = (const int*)d_in[2];
  const float* Wq   = (const float*)d_in[3];
  const float* Wk   = (const float*)d_in[4];
  const float* Wv   = (const float*)d_in[5];
  const float* Wo   = (const float*)d_in[6];
  const float* bo   = (const float*)d_in[7];
  const float* W1   = (const float*)d_in[8];
  const float* b1   = (const float*)d_in[9];
  const float* W2   = (const float*)d_in[10];
  const float* b2   = (const float*)d_in[11];
  const float* bn1w = (const float*)d_in[12];
  const float* bn1b = (const float*)d_in[13];
  const float* bn2w = (const float*)d_in[14];
  const float* bn2b = (const float*)d_in[15];

  const int N = in_sizes[0] / DM;
  const int E = in_sizes[1];
  const float invN = 1.0f / (float)N;

  float* ws = (float*)d_ws;
  size_t off = 0;
  float* Q      = ws + off; off += (size_t)N * DM;
  float* Kp     = ws + off; off += (size_t)N * DM;
  float* Vp     = ws + off; off += (size_t)N * DM;
  float* xbuf   = ws + off; off += (size_t)N * DM;
  float* ypre   = ws + off; off += (size_t)N * DM;
  float* wV     = ws + off; off += (size_t)N * DM;   // zeroed region starts here
  float* zbuf   = ws + off; off += (size_t)N * 8;
  float* stats1 = ws + off; off += 256;
  float* stats2 = ws + off; off += 256;

  const int nodeBlocks = (N + 15) / 16;
  const int zeroCount  = N * DM + N * 8 + 512;

  zero_kernel<<<(zeroCount + 255) / 256, 256, 0, stream>>>(wV, zeroCount);

  qkv_kernel<<<nodeBlocks, 256, 0, stream>>>(h, Wq, Wk, Wv, Q, Kp, Vp, N);

  int edgeBlocks = (int)(((size_t)E * DM + 255) / 256);
  edge_attn_kernel<<<edgeBlocks, 256, 0, stream>>>(Q, Kp, Vp, src, dst, wV,
                                                   zbuf, E);

  attn_out_kernel<<<nodeBlocks, 256, 0, stream>>>(wV, zbuf, h, Wo, bo, xbuf, N);

  col_stats_kernel<<<(N + 255) / 256, 256, 0, stream>>>(xbuf, stats1, N);

  ffn_kernel<<<nodeBlocks, 256, 0, stream>>>(xbuf, stats1, bn1w, bn1b, W1, b1,
                                             W2, b2, ypre, invN, N);

  col_stats_kernel<<<(N + 255) / 256, 256, 0, stream>>>(ypre, stats2, N);

  bn_apply_kernel<<<(N * DM + 255) / 256, 256, 0, stream>>>(
      ypre, stats2, bn2w, bn2b, (float*)d_out, invN, N * DM);
}